// GeometryAwareLiDARProcessor_30356828848554
// MI455X (gfx1250) — compile-verified
//
#include <hip/hip_runtime.h>

#define HH 1024
#define WW 2048
#define GW (WW + 2)                 // 2050
#define GH (HH + 2)                 // 1026
#define GRID_ELEMS (GH * GW)        // 2,103,300 floats (~8.4 MB)
#define C_OUT 64
#define BN_EPS 1e-5f
#define STAGE_STRIDE 68             // 64 + 4 pad: removes half-wave bank conflict

typedef __attribute__((ext_vector_type(2)))  float v2f;
typedef __attribute__((ext_vector_type(8)))  float v8f;

// ---------------- zero workspace (grid + stats accumulators) ----------------
__global__ void k_zero(float* __restrict__ p, long long n) {
  long long i = (long long)blockIdx.x * blockDim.x + threadIdx.x;
  long long stride = (long long)gridDim.x * blockDim.x;
  for (; i < n; i += stride) p[i] = 0.0f;
}

// ---------------- scatter active features into padded dense grid -----------
__global__ void k_scatter(const float* __restrict__ feats,
                          const int* __restrict__ coords,
                          float* __restrict__ grid, int N) {
  int n = blockIdx.x * blockDim.x + threadIdx.x;
  if (n < N) {
    int y = coords[2 * n] + 1;
    int x = coords[2 * n + 1] + 1;
    grid[y * GW + x] = feats[n];
  }
}

// ---------------- stats: column sums s[9] and Gram G[45] (upper tri) -------
__global__ void k_stats(const float* __restrict__ grid,
                        const int* __restrict__ coords,
                        float* __restrict__ sG, int N) {
  float s[9];
  float G[45];
#pragma unroll
  for (int k = 0; k < 9; k++) s[k] = 0.0f;
#pragma unroll
  for (int t = 0; t < 45; t++) G[t] = 0.0f;

  int i = blockIdx.x * blockDim.x + threadIdx.x;
  int stride = gridDim.x * blockDim.x;
  for (int n = i; n < N; n += stride) {
    int y = coords[2 * n] + 1;
    int x = coords[2 * n + 1] + 1;
    float v[9];
#pragma unroll
    for (int k = 0; k < 9; k++)
      v[k] = grid[(y + k / 3 - 1) * GW + (x + k % 3 - 1)];
    int t = 0;
#pragma unroll
    for (int k = 0; k < 9; k++) {
      s[k] += v[k];
#pragma unroll
      for (int j = k; j < 9; j++) G[t++] += v[k] * v[j];
    }
  }

  __shared__ float red[54];
  if (threadIdx.x < 54) red[threadIdx.x] = 0.0f;
  __syncthreads();
#pragma unroll
  for (int k = 0; k < 9; k++) atomicAdd(&red[k], s[k]);
#pragma unroll
  for (int t = 0; t < 45; t++) atomicAdd(&red[9 + t], G[t]);
  __syncthreads();
  if (threadIdx.x < 54) atomicAdd(&sG[threadIdx.x], red[threadIdx.x]);
}

// ---------------- fold BN into weights: w' = w*scale, shift = beta-mean*scale
__global__ void k_fold(const float* __restrict__ sG,
                       const float* __restrict__ weight,  // [9][1][64]
                       const float* __restrict__ gamma,
                       const float* __restrict__ beta,
                       float* __restrict__ wfold,         // [16][64] (rows 9..15 zero)
                       float* __restrict__ shift,         // [64]
                       int N) {
  int c = threadIdx.x;
  if (c >= C_OUT) return;
  float invN = 1.0f / (float)N;
  float w[9];
#pragma unroll
  for (int k = 0; k < 9; k++) w[k] = weight[k * C_OUT + c];

  float mean = 0.0f;
#pragma unroll
  for (int k = 0; k < 9; k++) mean += sG[k] * w[k];
  mean *= invN;

  float ex2 = 0.0f;
  int t = 9;
#pragma unroll
  for (int k = 0; k < 9; k++) {
#pragma unroll
    for (int j = k; j < 9; j++) {
      float g = sG[t++];
      ex2 += ((j == k) ? 1.0f : 2.0f) * g * w[k] * w[j];
    }
  }
  ex2 *= invN;

  float var = ex2 - mean * mean;
  float scale = gamma[c] * rsqrtf(var + BN_EPS);
  shift[c] = beta[c] - mean * scale;
#pragma unroll
  for (int k = 0; k < 16; k++)
    wfold[k * C_OUT + c] = (k < 9) ? (w[k] * scale) : 0.0f;
}

// ---------------- main pass: gather -> WMMA GEMM -> bias/ReLU -> b128 store
__global__ void __launch_bounds__(256)
k_conv_wmma(const float* __restrict__ grid,
            const int* __restrict__ coords,
            const float* __restrict__ wfold,
            const float* __restrict__ shift,
            float* __restrict__ out, int N) {
  __shared__ float wlds[16 * C_OUT];                 // folded weights, K pad 16
  __shared__ float shlds[C_OUT];
  __shared__ float stage[8 * 16 * STAGE_STRIDE];     // per-wave 16x64 tile staging
  for (int i = threadIdx.x; i < 16 * C_OUT; i += blockDim.x) wlds[i] = wfold[i];
  if (threadIdx.x < C_OUT) shlds[threadIdx.x] = shift[threadIdx.x];
  __syncthreads();

  const int lane = threadIdx.x & 31;   // wave32
  const int half = lane >> 4;          // 0 or 1
  const int row  = lane & 15;
  const int col  = lane & 15;
  const int waveInBlock   = threadIdx.x >> 5;
  const int wavesPerBlock = blockDim.x >> 5;
  const int waveGlobal = blockIdx.x * wavesPerBlock + waveInBlock;
  const int totalWaves = gridDim.x * wavesPerBlock;
  const int numTiles = (N + 15) / 16;
  float* mystage = stage + waveInBlock * (16 * STAGE_STRIDE);

  // Tile-invariant B fragments (4 K-steps x 4 N-tiles) and bias, hoisted.
  v2f bfrag[4][4];
  float shv[4];
#pragma unroll
  for (int ct = 0; ct < 4; ct++) {
    shv[ct] = shlds[ct * 16 + col];
#pragma unroll
    for (int t = 0; t < 4; t++) {
      int kb = 4 * t + 2 * half;
      v2f b;
      b.x = wlds[(kb + 0) * C_OUT + ct * 16 + col];
      b.y = wlds[(kb + 1) * C_OUT + ct * 16 + col];
      bfrag[ct][t] = b;
    }
  }

  for (int tile = waveGlobal; tile < numTiles; tile += totalWaves) {
    const int base = tile * 16;
    int n = base + row;
    int nc = (n < N) ? n : (N - 1);
    float vmask = (n < N) ? 1.0f : 0.0f;
    int y = coords[2 * nc] + 1;
    int x = coords[2 * nc + 1] + 1;

    // A fragments: 16x4 f32 per K-step; lane holds K = 4*t + 2*half + {0,1}
    v2f a[4];
#pragma unroll
    for (int t = 0; t < 4; t++) {
      int k0 = 4 * t + 2 * half;
      float e0 = (k0 + 0 < 9) ? grid[(y + (k0 + 0) / 3 - 1) * GW + (x + (k0 + 0) % 3 - 1)] : 0.0f;
      float e1 = (k0 + 1 < 9) ? grid[(y + (k0 + 1) / 3 - 1) * GW + (x + (k0 + 1) % 3 - 1)] : 0.0f;
      v2f av; av.x = e0 * vmask; av.y = e1 * vmask;
      a[t] = av;
    }

    // GEMM + bias + ReLU, staged into LDS (padded stride: conflict-free)
#pragma unroll
    for (int ct = 0; ct < 4; ct++) {
      v8f acc = {};
#pragma unroll
      for (int t = 0; t < 4; t++) {
        acc = __builtin_amdgcn_wmma_f32_16x16x4_f32(
            false, a[t], false, bfrag[ct][t], (short)0, acc, false, false);
      }
      int c = ct * 16 + col;
      float sh = shv[ct];
#pragma unroll
      for (int v = 0; v < 8; v++) {
        int m = v + 8 * half;
        float r = acc[v] + sh;
        mystage[m * STAGE_STRIDE + c] = (r > 0.0f) ? r : 0.0f;
      }
    }

    // Read back row-major, write 512B-contiguous b128 stores.
    // Tile output is out[base*64 .. (base+16)*64): 1024 consecutive floats.
#pragma unroll
    for (int r = 0; r < 8; r++) {
      int fi = r * 128 + lane * 4;       // float index within 1024-float tile
      int rowInTile = fi >> 6;           // = 2r + (lane>>4)
      int cst = fi & 63;
      int nn = base + rowInTile;
      if (nn < N) {
        const float4* src = (const float4*)&mystage[rowInTile * STAGE_STRIDE + cst];
        float4* dst = (float4*)&out[(long long)nn * C_OUT + cst];
        *dst = *src;
      }
    }
  }
}

// ---------------------------------------------------------------------------
extern "C" void kernel_launch(void* const* d_in, const int* in_sizes, int n_in,
                              void* d_out, int out_size, void* d_ws, size_t ws_size,
                              hipStream_t stream) {
  const float* feats  = (const float*)d_in[0];   // [N,1]
  const float* weight = (const float*)d_in[1];   // [9,1,64]
  const float* gamma  = (const float*)d_in[2];   // [64]
  const float* beta   = (const float*)d_in[3];   // [64]
  const int*   coords = (const int*)d_in[4];     // [N,2]
  float* out = (float*)d_out;                    // [N,64]
  const int N = in_sizes[0];                     // C_IN == 1

  float* ws     = (float*)d_ws;
  float* grid   = ws;                            // GRID_ELEMS floats
  float* sG     = ws + GRID_ELEMS;               // 54 used (padded to 64)
  float* wfold  = sG + 64;                       // 16*64 floats
  float* shiftp = wfold + 16 * C_OUT;            // 64 floats

  long long zn = (long long)GRID_ELEMS + 64;     // grid + stats accumulators
  k_zero<<<2048, 256, 0, stream>>>(ws, zn);
  k_scatter<<<(N + 255) / 256, 256, 0, stream>>>(feats, coords, grid, N);
  k_stats<<<1024, 256, 0, stream>>>(grid, coords, sG, N);
  k_fold<<<1, 64, 0, stream>>>(sG, weight, gamma, beta, wfold, shiftp, N);

  int numTiles = (N + 15) / 16;
  int blocks = (numTiles + 7) / 8;               // 8 waves/block, 1 tile/wave
  if (blocks < 1) blocks = 1;
  k_conv_wmma<<<blocks, 256, 0, stream>>>(grid, coords, wfold, shiftp, out, N);
}